// ColBertModel_14628658610754
// MI455X (gfx1250) — compile-verified
//
#include <hip/hip_runtime.h>
#include <hip/hip_bf16.h>
#include <float.h>

// ---------------------------------------------------------------------------
// N=4096 tokens, E=768, P=1024, C=128 codebooks, K=256 codewords, D=6.
// d_out = pq_simi [N,P] | pq_centers [N,E] | codeword_assign [N,C,K] | quantized [N,E]
// ---------------------------------------------------------------------------

typedef float v2f __attribute__((ext_vector_type(2)));
typedef float v8f __attribute__((ext_vector_type(8)));

// ---------------------------------------------------------------------------
// Row sum-of-squares: out[row] = sum_j X[row,j]^2   (one 256-thread block/row)
// ---------------------------------------------------------------------------
__global__ __launch_bounds__(256) void rowsumsq_kernel(
    const float* __restrict__ X, float* __restrict__ out, int cols)
{
    const int row = blockIdx.x;
    const int t   = threadIdx.x;
    __shared__ float sp[256];
    float s = 0.f;
    for (int j = t; j < cols; j += 256) {
        float x = X[(size_t)row * cols + j];
        s += x * x;
    }
    sp[t] = s;
    __syncthreads();
    for (int off = 128; off > 0; off >>= 1) {
        if (t < off) sp[t] += sp[t + off];
        __syncthreads();
    }
    if (t == 0) out[row] = sp[0];
}

// ---------------------------------------------------------------------------
// Fp32 WMMA GEMM:  C = A * B^T     (A [M,K] rm, B [Ncols,K] rm)
// Block tile 64(M) x 128(N), 256 threads = 8 waves in a 2x4 grid; each wave
// register-blocks 2x2 WMMA tiles (32x32 output, 4 v8f accumulators) so each
// A/B fragment pair feeds 4 v_wmma_f32_16x16x4_f32 ops (halves LDS traffic,
// 4 independent accumulation chains for ILP between ds_load waits).
//
// Fragment layouts per CDNA5 ISA 7.12.2 (32-bit, wave32):
//   A 16x4 : lane = (k>=2)*16 + m , vgpr = k&1
//   B 4x16 : lane = (k>=2)*16 + n , vgpr = k&1
//   C 16x16: vgpr g -> M = g + 8*(lane>=16), N = lane&15
//
// simi_mode: write 2*acc - n1[row] - n2[col]  (pq_simi epilogue fusion)
// ---------------------------------------------------------------------------
__global__ __launch_bounds__(256) void gemm_nt_wmma_kernel(
    const float* __restrict__ A,
    const float* __restrict__ B,
    float* __restrict__ Cout,
    const float* __restrict__ n1,
    const float* __restrict__ n2,
    int M, int Ncols, int K, int simi_mode)
{
    __shared__ float lA[32][65];    // [k][m], +1 pad: conflict-free both ways
    __shared__ float lB[32][129];   // [k][n], +1 pad

    const int t    = threadIdx.x;
    const int m0   = blockIdx.y * 64;
    const int n0   = blockIdx.x * 128;
    const int w    = t >> 5;            // wave 0..7
    const int lane = t & 31;
    const int wr   = w >> 2;            // 0..1 -> 32-row strip
    const int wc   = w & 3;             // 0..3 -> 32-col strip
    const int lm   = lane & 15;
    const int hk   = (lane >> 4) << 1;  // K offset 0 or 2 per lane half
    const int mb   = wr * 32;
    const int nb   = wc * 32;

    v8f acc00 = {0.f,0.f,0.f,0.f,0.f,0.f,0.f,0.f};
    v8f acc01 = acc00, acc10 = acc00, acc11 = acc00;

    const int ldr = t >> 3;             // 0..31
    const int ldc = (t & 7) << 2;       // 0,4,...,28

    for (int kk = 0; kk < K; kk += 32) {
        // A tile 64x32 -> lA[k][m]   (2 float4 per thread, coalesced on K)
        #pragma unroll
        for (int rep = 0; rep < 2; ++rep) {
            const int ar = rep * 32 + ldr;
            const float4 va = *(const float4*)(A + (size_t)(m0 + ar) * K + kk + ldc);
            lA[ldc + 0][ar] = va.x;
            lA[ldc + 1][ar] = va.y;
            lA[ldc + 2][ar] = va.z;
            lA[ldc + 3][ar] = va.w;
        }
        // B tile 128x32 -> lB[k][n]  (4 float4 per thread)
        #pragma unroll
        for (int rep = 0; rep < 4; ++rep) {
            const int br = rep * 32 + ldr;
            const float4 vb = *(const float4*)(B + (size_t)(n0 + br) * K + kk + ldc);
            lB[ldc + 0][br] = vb.x;
            lB[ldc + 1][br] = vb.y;
            lB[ldc + 2][br] = vb.z;
            lB[ldc + 3][br] = vb.w;
        }
        __syncthreads();

        #pragma unroll
        for (int kb = 0; kb < 8; ++kb) {         // 8 x (K=4) per 32-chunk
            const int k0 = (kb << 2) + hk;
            v2f a0, a1, b0, b1;
            a0[0] = lA[k0 + 0][mb + lm];       a0[1] = lA[k0 + 1][mb + lm];
            a1[0] = lA[k0 + 0][mb + 16 + lm];  a1[1] = lA[k0 + 1][mb + 16 + lm];
            b0[0] = lB[k0 + 0][nb + lm];       b0[1] = lB[k0 + 1][nb + lm];
            b1[0] = lB[k0 + 0][nb + 16 + lm];  b1[1] = lB[k0 + 1][nb + 16 + lm];
            acc00 = __builtin_amdgcn_wmma_f32_16x16x4_f32(false, a0, false, b0, (short)0, acc00, false, false);
            acc01 = __builtin_amdgcn_wmma_f32_16x16x4_f32(false, a0, false, b1, (short)0, acc01, false, false);
            acc10 = __builtin_amdgcn_wmma_f32_16x16x4_f32(false, a1, false, b0, (short)0, acc10, false, false);
            acc11 = __builtin_amdgcn_wmma_f32_16x16x4_f32(false, a1, false, b1, (short)0, acc11, false, false);
        }
        __syncthreads();
    }

    // Epilogue: 4 tiles per wave, C/D layout as documented above
    const int rhalf = (lane >> 4) << 3;
    #pragma unroll
    for (int i = 0; i < 2; ++i) {
        const int rbase = m0 + mb + i * 16 + rhalf;
        #pragma unroll
        for (int j = 0; j < 2; ++j) {
            const int col = n0 + nb + j * 16 + lm;
            const v8f& acc = (i == 0) ? (j == 0 ? acc00 : acc01)
                                      : (j == 0 ? acc10 : acc11);
            if (simi_mode) {
                const float nn2 = n2[col];
                #pragma unroll
                for (int g = 0; g < 8; ++g) {
                    const int row = rbase + g;
                    Cout[(size_t)row * Ncols + col] = 2.0f * acc[g] - n1[row] - nn2;
                }
            } else {
                #pragma unroll
                for (int g = 0; g < 8; ++g) {
                    const int row = rbase + g;
                    Cout[(size_t)row * Ncols + col] = acc[g];
                }
            }
        }
    }
}

// ---------------------------------------------------------------------------
// Per-token argmax over pq_simi row (first-max tie-break like jnp.argmax),
// then gather pq_centers row. One 256-thread block per token.
// ---------------------------------------------------------------------------
__global__ __launch_bounds__(256) void argmax_gather_kernel(
    const float* __restrict__ simi,       // [N,P]
    const float* __restrict__ pq_assign,  // [P,E]
    float* __restrict__ pq_centers,       // [N,E]
    int* __restrict__ pq_index,           // [N]
    int P, int E)
{
    const int n = blockIdx.x;
    const int t = threadIdx.x;
    __shared__ float sv[256];
    __shared__ int   si[256];

    float best = -FLT_MAX;
    int   bidx = P;
    for (int p = t; p < P; p += 256) {
        float val = simi[(size_t)n * P + p];
        if (val > best || (val == best && p < bidx)) { best = val; bidx = p; }
    }
    sv[t] = best; si[t] = bidx;
    __syncthreads();
    for (int off = 128; off > 0; off >>= 1) {
        if (t < off) {
            float v2 = sv[t + off]; int i2 = si[t + off];
            if (v2 > sv[t] || (v2 == sv[t] && i2 < si[t])) { sv[t] = v2; si[t] = i2; }
        }
        __syncthreads();
    }
    const int idx = si[0];
    if (t == 0) pq_index[n] = idx;
    for (int e = t; e < E; e += 256)
        pq_centers[(size_t)n * E + e] = pq_assign[(size_t)idx * E + e];
}

// ---------------------------------------------------------------------------
// Code selection: one wave32 per (token n, codebook c); 8 codewords per lane.
//   r = v - pq_centers (6 floats); proba_k = -|r - cb[c,k]|^2
//   softmax(proba/0.01) -> 256 coalesced floats; STE forward one-hot gather.
// Pure shuffle reductions, zero barriers. 524288 waves of parallelism; this
// stage is store-bound (536 MB of codeword_assign).
// ---------------------------------------------------------------------------
__global__ __launch_bounds__(256) void code_select_kernel(
    const float* __restrict__ v,        // [N,768]
    const float* __restrict__ pqc,      // [N,768]
    const float* __restrict__ cb,       // [128,256,6]
    float* __restrict__ assign_out,     // [N,128,256]
    float* __restrict__ outvec)         // [N,768]
{
    const int lane = threadIdx.x & 31;
    const int pair = blockIdx.x * 8 + (threadIdx.x >> 5);  // n*128 + c
    const int n = pair >> 7;
    const int c = pair & 127;
    const size_t vbase = (size_t)n * 768 + c * 6;

    // residual, identical across the wave (broadcast loads hit cache)
    float r[6];
    #pragma unroll
    for (int d = 0; d < 6; ++d)
        r[d] = v[vbase + d] - pqc[vbase + d];

    // distances: lane owns k = j*32 + lane, j = 0..7 (k ascending within lane)
    float p[8];
    float best = -FLT_MAX;
    int   bk   = 256;
    #pragma unroll
    for (int j = 0; j < 8; ++j) {
        const int k = j * 32 + lane;
        const float* wv = cb + ((size_t)c * 256 + k) * 6;
        float s = 0.f;
        #pragma unroll
        for (int d = 0; d < 6; ++d) {
            const float dd = r[d] - wv[d];
            s += dd * dd;
        }
        p[j] = -s;
        if (p[j] > best) { best = p[j]; bk = k; }   // strict > keeps first
    }
    // wave max + first-argmax (min index on ties)
    #pragma unroll
    for (int off = 16; off > 0; off >>= 1) {
        const float ov = __shfl_xor(best, off, 32);
        const int   oi = __shfl_xor(bk,   off, 32);
        if (ov > best || (ov == best && oi < bk)) { best = ov; bk = oi; }
    }

    // softmax with T=0.01 (exp arg <= 0)
    float e[8];
    float sum = 0.f;
    #pragma unroll
    for (int j = 0; j < 8; ++j) {
        e[j] = __expf((p[j] - best) * 100.0f);
        sum += e[j];
    }
    #pragma unroll
    for (int off = 16; off > 0; off >>= 1)
        sum += __shfl_xor(sum, off, 32);
    const float inv = 1.0f / sum;

    float* dst = assign_out + (size_t)pair * 256;
    #pragma unroll
    for (int j = 0; j < 8; ++j)
        dst[j * 32 + lane] = e[j] * inv;            // 128B coalesced per step

    // STE forward: hard one-hot gather
    if (lane < 6)
        outvec[vbase + lane] = pqc[vbase + lane] + cb[((size_t)c * 256 + bk) * 6 + lane];
}

// ---------------------------------------------------------------------------
// quantized = outvec / max(||outvec||, 1e-12), one block per token
// ---------------------------------------------------------------------------
__global__ __launch_bounds__(256) void normalize_kernel(
    const float* __restrict__ outvec, float* __restrict__ qout, int E)
{
    const int n = blockIdx.x;
    const int t = threadIdx.x;
    __shared__ float sp[256];
    float s = 0.f;
    for (int e = t; e < E; e += 256) {
        float x = outvec[(size_t)n * E + e];
        s += x * x;
    }
    sp[t] = s;
    __syncthreads();
    for (int off = 128; off > 0; off >>= 1) {
        if (t < off) sp[t] += sp[t + off];
        __syncthreads();
    }
    const float norm = fmaxf(sqrtf(sp[0]), 1e-12f);
    const float inv  = 1.0f / norm;
    for (int e = t; e < E; e += 256)
        qout[(size_t)n * E + e] = outvec[(size_t)n * E + e] * inv;
}

// ---------------------------------------------------------------------------
// Launch
// ---------------------------------------------------------------------------
extern "C" void kernel_launch(void* const* d_in, const int* in_sizes, int n_in,
                              void* d_out, int out_size, void* d_ws, size_t ws_size,
                              hipStream_t stream) {
    const float* vectors   = (const float*)d_in[0];  // [N,E]
    const float* rotate    = (const float*)d_in[1];  // [E,E]
    const float* pq_assign = (const float*)d_in[2];  // [P,E]
    const float* codebooks = (const float*)d_in[3];  // [C,K,D]

    const int E = 768;
    const int N = in_sizes[0] / E;   // 4096
    const int P = in_sizes[2] / E;   // 1024
    const int C = 128;

    // d_out sections (fp32, concatenated in return order)
    float* out_simi   = (float*)d_out;                       // [N,P]
    float* out_pqc    = out_simi + (size_t)N * P;            // [N,E]
    float* out_assign = out_pqc  + (size_t)N * E;            // [N,128,256]
    float* out_quant  = out_assign + (size_t)N * 128 * 256;  // [N,E]

    // workspace layout
    char*  ws   = (char*)d_ws;
    float* v_ws = (float*)ws;   ws += (size_t)N * E * sizeof(float);
    float* n1   = (float*)ws;   ws += (size_t)N * sizeof(float);
    float* n2   = (float*)ws;   ws += (size_t)P * sizeof(float);
    float* ovec = (float*)ws;   ws += (size_t)N * E * sizeof(float);
    int*   pidx = (int*)ws;

    // n2 = row sumsq of pq_assign
    rowsumsq_kernel<<<P, 256, 0, stream>>>(pq_assign, n2, E);

    // v = vectors @ rotate^T          (fp32 WMMA, 64x128 block tile)
    {
        dim3 grid(E / 128, N / 64);
        gemm_nt_wmma_kernel<<<grid, 256, 0, stream>>>(
            vectors, rotate, v_ws, nullptr, nullptr, N, E, E, 0);
    }

    // n1 = row sumsq of v
    rowsumsq_kernel<<<N, 256, 0, stream>>>(v_ws, n1, E);

    // pq_simi = 2 * (v @ pq_assign^T) - n1 - n2   (fp32 WMMA, fused epilogue)
    {
        dim3 grid(P / 128, N / 64);
        gemm_nt_wmma_kernel<<<grid, 256, 0, stream>>>(
            v_ws, pq_assign, out_simi, n1, n2, N, P, E, 1);
    }

    // pq_index / pq_centers
    argmax_gather_kernel<<<N, 256, 0, stream>>>(
        out_simi, pq_assign, out_pqc, pidx, P, E);

    // codeword_assign + STE-forward quantized accumulation (1 wave per (n,c))
    code_select_kernel<<<(N * C) / 8, 256, 0, stream>>>(
        v_ws, out_pqc, codebooks, out_assign, ovec);

    // quantized_vecs = normalize(pq_centers + q)
    normalize_kernel<<<N, 256, 0, stream>>>(ovec, out_quant, E);
}